// Bi_Attention_90950227460841
// MI455X (gfx1250) — compile-verified
//
#include <hip/hip_runtime.h>
#include <hip/hip_bf16.h>

typedef float v2f __attribute__((ext_vector_type(2)));
typedef float v8f __attribute__((ext_vector_type(8)));

#define HH 512
#define WW 512
#define CC 32
#define BB 4
#define NN 50000
#define HWW (HH * WW)
#define MM (BB * NN)   // 200000 rows, divisible by 16

// ---------------------------------------------------------------- fill kernels
__global__ void fill_zero_f4(float4* __restrict__ p, size_t n4) {
  size_t i = (size_t)blockIdx.x * blockDim.x + threadIdx.x;
  size_t stride = (size_t)gridDim.x * blockDim.x;
  float4 z = make_float4(0.f, 0.f, 0.f, 0.f);
  for (; i < n4; i += stride) p[i] = z;
}

__global__ void fill_i32(int* __restrict__ p, int v, size_t n) {
  size_t i = (size_t)blockIdx.x * blockDim.x + threadIdx.x;
  size_t stride = (size_t)gridDim.x * blockDim.x;
  for (; i < n; i += stride) p[i] = v;
}

// ---------------------------------------------------------------- grid build
__global__ void build_grid(const int* __restrict__ coors, int* __restrict__ grid) {
  int t = blockIdx.x * blockDim.x + threadIdx.x;
  if (t >= MM) return;
  int b = t / NN;
  int i = t - b * NN;
  int y = coors[(size_t)t * 2 + 0];
  int x = coors[(size_t)t * 2 + 1];
  grid[(size_t)b * HWW + y * WW + x] = i;
}

// ---------------------------------------------------------------- WMMA projection
// out[M x NC] = X[M x 32] @ W^T + bias, where W rows are wqkv[woff .. woff+NC)
// (row-major [*,32]) and bias[n] = bqkv[woff+n]. Exact fp32 via
// V_WMMA_F32_16X16X4_F32: one wave per 16-row tile, 8 K-steps per 16x16 tile.
__global__ void proj_wmma(const float* __restrict__ X,
                          const float* __restrict__ wqkv,
                          const float* __restrict__ bqkv,
                          int woff, int NC,
                          float* __restrict__ out) {
  int wave = (int)((blockIdx.x * blockDim.x + threadIdx.x) >> 5);
  int lane = threadIdx.x & 31;
  const int numTiles = MM >> 4;            // wave-uniform guard: EXEC stays all-1s
  if (wave >= numTiles) return;

  const int row0 = wave << 4;
  const int lr   = lane & 15;
  const int ksel = (lane < 16) ? 0 : 2;    // A/B K-split per ISA f32 layout
  const int rsel = (lane < 16) ? 0 : 8;    // C/D row split

  const float* xr = X + (size_t)(row0 + lr) * 32 + ksel;

  for (int n0 = 0; n0 < NC; n0 += 16) {
    const int ncol = n0 + lr;
    const float* wr = wqkv + (size_t)(woff + ncol) * 32 + ksel;
    v8f acc = {};
#pragma unroll
    for (int k0 = 0; k0 < 32; k0 += 4) {
      v2f a, bf;
      a[0]  = xr[k0];     a[1]  = xr[k0 + 1];      // A[row][k0+ksel], A[row][k0+ksel+1]
      bf[0] = wr[k0];     bf[1] = wr[k0 + 1];      // B[k0+ksel][ncol], B[k0+ksel+1][ncol]
      acc = __builtin_amdgcn_wmma_f32_16x16x4_f32(
          /*neg_a=*/false, a, /*neg_b=*/false, bf,
          /*c_mod=*/(short)0, acc, /*reuse_a=*/false, /*reuse_b=*/false);
    }
    const float bias = bqkv[woff + ncol];
#pragma unroll
    for (int r = 0; r < 8; r++) {
      out[(size_t)(row0 + rsel + r) * NC + ncol] = acc[r] + bias;
    }
  }
}

// ---------------------------------------------------------------- fused attention
// One thread per query voxel: gather 9 projected K/V rows (bias row if empty),
// 2-head softmax over 9, out-projection (LDS-staged w_out), residual, scatter.
__global__ void attn_scatter(const float* __restrict__ qf,    // [B*N,32] original feats
                             const int*   __restrict__ qc,    // [B*N,2]
                             const float* __restrict__ Qp,    // [B*N,32]
                             const float* __restrict__ KVp,   // [B*N,64] (K||V)
                             const int*   __restrict__ grid,  // [B,HW]
                             const float* __restrict__ bqkv,  // [96]
                             const float* __restrict__ wout,  // [32,32]
                             const float* __restrict__ bout,  // [32]
                             float* __restrict__ canvas) {    // [B,32,H,W]
  __shared__ float sW[CC * CC];
  __shared__ float sBo[CC], sBk[CC], sBv[CC];
  for (int i = threadIdx.x; i < CC * CC; i += blockDim.x) sW[i] = wout[i];
  if (threadIdx.x < CC) {
    sBo[threadIdx.x] = bout[threadIdx.x];
    sBk[threadIdx.x] = bqkv[32 + threadIdx.x];
    sBv[threadIdx.x] = bqkv[64 + threadIdx.x];
  }
  __syncthreads();

  int t = blockIdx.x * blockDim.x + threadIdx.x;
  if (t >= MM) return;
  int b = t / NN;
  int y = qc[(size_t)t * 2 + 0];
  int x = qc[(size_t)t * 2 + 1];

  const int SY[9] = {0, -1, 1, 0, -1, 1, 0, -1, 1};
  const int SX[9] = {0,  0, 0, 1,  1, 1, -1, -1, -1};

  float q[32];
  {
    const float* qpr = Qp + (size_t)t * 32;
#pragma unroll
    for (int d = 0; d < 32; d += 4) {
      float4 v = *(const float4*)(qpr + d);
      q[d] = v.x; q[d + 1] = v.y; q[d + 2] = v.z; q[d + 3] = v.w;
    }
  }

  int   nidx[9];
  float s0[9], s1[9];
  const int* gb = grid + (size_t)b * HWW;

#pragma unroll
  for (int k = 0; k < 9; k++) {
    int ny = y + SY[k], nx = x + SX[k];
    int idx = -1;
    if (ny >= 0 && ny < HH && nx >= 0 && nx < WW) idx = gb[ny * WW + nx];
    nidx[k] = idx;
    float a0 = 0.f, a1 = 0.f;
    if (idx >= 0) {
      const float* kp = KVp + ((size_t)b * NN + idx) * 64;
#pragma unroll
      for (int d = 0; d < 16; d += 4) {
        float4 kv = *(const float4*)(kp + d);
        a0 += q[d] * kv.x + q[d + 1] * kv.y + q[d + 2] * kv.z + q[d + 3] * kv.w;
        float4 kv2 = *(const float4*)(kp + 16 + d);
        a1 += q[16 + d] * kv2.x + q[17 + d] * kv2.y + q[18 + d] * kv2.z + q[19 + d] * kv2.w;
      }
    } else {  // empty cell: key input was zeroed -> projected K = bk
#pragma unroll
      for (int d = 0; d < 16; d++) {
        a0 += q[d] * sBk[d];
        a1 += q[16 + d] * sBk[16 + d];
      }
    }
    s0[k] = a0 * 0.25f;   // 1/sqrt(HD), HD=16
    s1[k] = a1 * 0.25f;
  }

  // softmax per head over 9 (all positions participate, matching reference)
  float m0 = s0[0], m1 = s1[0];
#pragma unroll
  for (int k = 1; k < 9; k++) { m0 = fmaxf(m0, s0[k]); m1 = fmaxf(m1, s1[k]); }
  float sum0 = 0.f, sum1 = 0.f;
#pragma unroll
  for (int k = 0; k < 9; k++) {
    s0[k] = __expf(s0[k] - m0); sum0 += s0[k];
    s1[k] = __expf(s1[k] - m1); sum1 += s1[k];
  }
  float r0 = 1.f / sum0, r1 = 1.f / sum1;

  float o[32];
#pragma unroll
  for (int d = 0; d < 32; d++) o[d] = 0.f;

#pragma unroll
  for (int k = 0; k < 9; k++) {
    float a0 = s0[k] * r0, a1 = s1[k] * r1;
    int idx = nidx[k];
    if (idx >= 0) {
      const float* vp = KVp + ((size_t)b * NN + idx) * 64 + 32;
#pragma unroll
      for (int d = 0; d < 16; d += 4) {
        float4 v = *(const float4*)(vp + d);
        o[d] += a0 * v.x; o[d + 1] += a0 * v.y; o[d + 2] += a0 * v.z; o[d + 3] += a0 * v.w;
        float4 v2 = *(const float4*)(vp + 16 + d);
        o[16 + d] += a1 * v2.x; o[17 + d] += a1 * v2.y; o[18 + d] += a1 * v2.z; o[19 + d] += a1 * v2.w;
      }
    } else {  // projected V of zeroed key = bv
#pragma unroll
      for (int d = 0; d < 16; d++) {
        o[d]      += a0 * sBv[d];
        o[16 + d] += a1 * sBv[16 + d];
      }
    }
  }

  // out-projection + residual + scatter to [C,H,W] canvas
  const float* qrow = qf + (size_t)t * 32;
  float* cv = canvas + (size_t)b * CC * HWW + (size_t)y * WW + x;
#pragma unroll
  for (int c = 0; c < CC; c++) {
    float acc = sBo[c];
#pragma unroll
    for (int d = 0; d < 32; d++) acc += o[d] * sW[c * 32 + d];
    cv[(size_t)c * HWW] = acc + qrow[c];
  }
}

// ---------------------------------------------------------------- launch
extern "C" void kernel_launch(void* const* d_in, const int* in_sizes, int n_in,
                              void* d_out, int out_size, void* d_ws, size_t ws_size,
                              hipStream_t stream) {
  const float* li_feats = (const float*)d_in[0];
  const int*   li_coors = (const int*)  d_in[1];
  const float* ra_feats = (const float*)d_in[2];
  const int*   ra_coors = (const int*)  d_in[3];
  const float* w_qkv1 = (const float*)d_in[4];
  const float* b_qkv1 = (const float*)d_in[5];
  const float* w_out1 = (const float*)d_in[6];
  const float* b_out1 = (const float*)d_in[7];
  const float* w_qkv2 = (const float*)d_in[8];
  const float* b_qkv2 = (const float*)d_in[9];
  const float* w_out2 = (const float*)d_in[10];
  const float* b_out2 = (const float*)d_in[11];

  float* out = (float*)d_out;

  // workspace layout: KVp [M,64] | Qp [M,32] | grid [B,HW]   (~81 MB)
  float* KVp  = (float*)d_ws;
  float* Qp   = KVp + (size_t)MM * 64;
  int*   grid = (int*)(Qp + (size_t)MM * 32);

  const size_t outElems = 2ull * BB * CC * HWW;   // both canvases
  fill_zero_f4<<<4096, 256, 0, stream>>>((float4*)out, outElems / 4);

  const int projBlocks = ((MM / 16) * 32 + 255) / 256;  // one wave per 16-row tile
  const int attnBlocks = (MM + 255) / 256;
  const int bgBlocks   = (MM + 255) / 256;

  for (int br = 0; br < 2; br++) {
    const float* qf   = br == 0 ? li_feats : ra_feats;
    const int*   qc   = br == 0 ? li_coors : ra_coors;
    const float* kvf  = br == 0 ? ra_feats : li_feats;
    const int*   kvc  = br == 0 ? ra_coors : li_coors;
    const float* wqkv = br == 0 ? w_qkv1 : w_qkv2;
    const float* bqkv = br == 0 ? b_qkv1 : b_qkv2;
    const float* wout = br == 0 ? w_out1 : w_out2;
    const float* bout = br == 0 ? b_out1 : b_out2;
    float* canvas = out + (size_t)br * BB * CC * HWW;

    fill_i32<<<1024, 256, 0, stream>>>(grid, -1, (size_t)BB * HWW);
    build_grid<<<bgBlocks, 256, 0, stream>>>(kvc, grid);

    // K||V projection of kv feats: rows 32..95 of w_qkv (wk then wv), NC=64
    proj_wmma<<<projBlocks, 256, 0, stream>>>(kvf, wqkv, bqkv, 32, 64, KVp);
    // Q projection of query feats: rows 0..31, NC=32
    proj_wmma<<<projBlocks, 256, 0, stream>>>(qf, wqkv, bqkv, 0, 32, Qp);

    attn_scatter<<<attnBlocks, 256, 0, stream>>>(qf, qc, Qp, KVp, grid,
                                                 bqkv, wout, bout, canvas);
  }
}